// BoltzmannUpdater_2370821947810
// MI455X (gfx1250) — compile-verified
//
#include <hip/hip_runtime.h>

#define QC     27
#define DTC    0.1f
#define TILE_E 128
#define BLOCK  256
#define TPB_T  4      // tiles per block in the pipelined edge kernel

#define AS1 __attribute__((address_space(1)))
#define AS3 __attribute__((address_space(3)))

typedef int v4i_vs __attribute__((__vector_size__(16)));

// ---- CDNA5 async global->LDS copy (ASYNCcnt path, ISA ch.8 §4) ----
__device__ __forceinline__ void async_b128(const void* g, void* l) {
#if defined(__gfx1250__) && __has_builtin(__builtin_amdgcn_global_load_async_to_lds_b128)
  __builtin_amdgcn_global_load_async_to_lds_b128(
      (AS1 v4i_vs*)(g), (AS3 v4i_vs*)(l), 0, 0);
#else
  *reinterpret_cast<float4*>(l) = *reinterpret_cast<const float4*>(g);
#endif
}

__device__ __forceinline__ void wait_async0() {
#if defined(__gfx1250__) && __has_builtin(__builtin_amdgcn_s_wait_asynccnt)
  __builtin_amdgcn_s_wait_asynccnt(0);
#else
  asm volatile("s_wait_asynccnt 0" ::: "memory");
#endif
}

// ---- Kernel 1: out = f + DT*(coll - src_term) (B128 vectorized); zero deg ----
__global__ __launch_bounds__(BLOCK) void k_init(
    const float* __restrict__ f, const float* __restrict__ coll,
    const float* __restrict__ srct, float* __restrict__ out,
    float* __restrict__ deg, int NQ, int N) {
  const float4* f4 = (const float4*)f;
  const float4* c4 = (const float4*)coll;
  const float4* s4 = (const float4*)srct;
  float4* o4 = (float4*)out;

  const int NQ4 = NQ >> 2;
  const int M   = (NQ4 > N) ? NQ4 : N;
  const int stride = gridDim.x * blockDim.x;
  const int gid = blockIdx.x * blockDim.x + threadIdx.x;

  for (int i = gid; i < M; i += stride) {
    if (i < NQ4) {
      float4 a = f4[i], c = c4[i], s = s4[i];
      float4 r;
      r.x = fmaf(DTC, c.x - s.x, a.x);
      r.y = fmaf(DTC, c.y - s.y, a.y);
      r.z = fmaf(DTC, c.z - s.z, a.z);
      r.w = fmaf(DTC, c.w - s.w, a.w);
      o4[i] = r;
    }
    if (i < N) deg[i] = 0.0f;
  }
  // scalar tail (NQ % 4 elements)
  const int base = NQ4 << 2;
  if (gid < NQ - base) {
    const int i = base + gid;
    out[i] = fmaf(DTC, coll[i] - srct[i], f[i]);
  }
}

// ---- Kernel 2: in-degree via f32 atomics, int4 streaming loads ----
__global__ __launch_bounds__(BLOCK) void k_degree(
    const int* __restrict__ dst, float* __restrict__ deg, int E) {
  const int4* d4 = (const int4*)dst;
  const int E4 = E >> 2;
  const int stride = gridDim.x * blockDim.x;
  const int gid = blockIdx.x * blockDim.x + threadIdx.x;

  for (int i = gid; i < E4; i += stride) {
    int4 d = d4[i];
    atomicAdd(&deg[d.x], 1.0f);
    atomicAdd(&deg[d.y], 1.0f);
    atomicAdd(&deg[d.z], 1.0f);
    atomicAdd(&deg[d.w], 1.0f);
  }
  const int base = E4 << 2;
  if (gid < E - base) atomicAdd(&deg[dst[base + gid]], 1.0f);
}

// ---- Kernel 3: double-buffered async-DMA pipeline over edge tiles ----
__global__ __launch_bounds__(BLOCK) void k_edges(
    const float* __restrict__ f, const float* __restrict__ xi,
    const float* __restrict__ w, const int* __restrict__ src,
    const int* __restrict__ dst, const float* __restrict__ deg,
    float* __restrict__ out, int E, int tiles) {
  __shared__ __align__(16) int   s_src[2][TILE_E];
  __shared__ __align__(16) int   s_dst[2][TILE_E];
  __shared__ __align__(16) float s_w[2][TILE_E];
  __shared__ float s_scale[2][TILE_E];
  __shared__ float s_xi[QC];

  const int tid = threadIdx.x;
  const int t0  = blockIdx.x * TPB_T;
  const int t1  = (t0 + TPB_T < tiles) ? (t0 + TPB_T) : tiles;
  if (t0 >= tiles) return;

  if (tid < QC) s_xi[tid] = xi[tid];

  // Issue async DMA of a FULL tile t into buffer b (512B x3, 16B/lane).
  auto issue = [&](int t, int b) {
    if ((t + 1) * TILE_E <= E) {
      if (tid < 96) {
        const int arr  = tid >> 5;   // 0=src, 1=dst, 2=w
        const int lane = tid & 31;
        const int e0   = t * TILE_E;
        const void* g;
        void* l;
        if (arr == 0)      { g = src + e0 + lane * 4; l = &s_src[b][lane * 4]; }
        else if (arr == 1) { g = dst + e0 + lane * 4; l = &s_dst[b][lane * 4]; }
        else               { g = w   + e0 + lane * 4; l = &s_w[b][lane * 4];   }
        async_b128(g, l);
      }
    }
  };

  issue(t0, 0);   // prologue

  for (int t = t0; t < t1; ++t) {
    const int b  = (t - t0) & 1;
    const int e0 = t * TILE_E;
    const int nE = (TILE_E < E - e0) ? TILE_E : (E - e0);

    wait_async0();                      // tile t's DMA (<=1 op/wave outstanding)
    if (nE < TILE_E) {                  // tail tile: guarded plain staging
      for (int i = tid; i < nE; i += BLOCK) {
        s_src[b][i] = src[e0 + i];
        s_dst[b][i] = dst[e0 + i];
        s_w[b][i]   = w[e0 + i];
      }
    }
    __syncthreads();                    // publish DMA'd/staged tile to all waves

    // Per-edge combined scale: -DT * w / deg[dst]
    for (int i = tid; i < nE; i += BLOCK)
      s_scale[b][i] = (-DTC) * s_w[b][i] / deg[s_dst[b][i]];
    __syncthreads();

    if (t + 1 < t1) issue(t + 1, b ^ 1);  // overlap next DMA with scatter below

    // Flat (edge, channel) items: contiguous 108B row gathers (L2-resident),
    // one device-scope f32 atomic per element into out.
    const int items = nE * QC;
    for (int i = tid; i < items; i += BLOCK) {
      const int e  = i / QC;            // constant-divisor -> mul/shift
      const int k  = i - e * QC;
      const int sN = s_src[b][e];
      const int dN = s_dst[b][e];
      const float diff = f[sN * QC + k] - f[dN * QC + k];
      atomicAdd(&out[dN * QC + k], diff * s_scale[b][e] * s_xi[k]);
    }
    __syncthreads();                    // WAR guard: buffer b reused at t+2
  }
}

extern "C" void kernel_launch(void* const* d_in, const int* in_sizes, int n_in,
                              void* d_out, int out_size, void* d_ws, size_t ws_size,
                              hipStream_t stream) {
  const float* f    = (const float*)d_in[0];
  const float* coll = (const float*)d_in[1];
  const float* srct = (const float*)d_in[2];
  const float* xi   = (const float*)d_in[3];
  const float* w    = (const float*)d_in[4];
  const int*   src  = (const int*)d_in[5];
  const int*   dst  = (const int*)d_in[6];
  float* out = (float*)d_out;
  float* deg = (float*)d_ws;   // N floats of scratch, zeroed each call in k_init

  const int Qn = in_sizes[3];                 // 27
  const int NQ = in_sizes[0];                 // N*Q
  const int N  = NQ / (Qn > 0 ? Qn : 1);
  const int E  = in_sizes[4];

  const int NQ4 = NQ >> 2;
  const int M   = (NQ4 > N) ? NQ4 : N;
  const int b0  = (M + BLOCK - 1) / BLOCK;
  k_init<<<b0, BLOCK, 0, stream>>>(f, coll, srct, out, deg, NQ, N);

  const int E4 = E >> 2;
  const int b1 = ((E4 > 0 ? E4 : 1) + BLOCK - 1) / BLOCK;
  k_degree<<<b1, BLOCK, 0, stream>>>(dst, deg, E);

  const int tiles = (E + TILE_E - 1) / TILE_E;
  const int b2    = (tiles + TPB_T - 1) / TPB_T;
  k_edges<<<b2, BLOCK, 0, stream>>>(f, xi, w, src, dst, deg, out, E, tiles);
}